// TripletLoss_10471130267652
// MI455X (gfx1250) — compile-verified
//
#include <hip/hip_runtime.h>
#include <math.h>

typedef float v2f __attribute__((ext_vector_type(2)));
typedef float v8f __attribute__((ext_vector_type(8)));

#define NB    512      // batch
#define FD    1024     // feature dim
#define MARGIN 0.2f
#define BIGV  1000000000.0f

// ---------------------------------------------------------------------------
// Kernel 1: per-row squared norms. One block per row, 256 threads.
// ---------------------------------------------------------------------------
__global__ __launch_bounds__(256)
void sqnorm_kernel(const float* __restrict__ F, float* __restrict__ sq) {
    __shared__ float red[256];
    const int row = blockIdx.x;
    const int tid = threadIdx.x;
    float acc = 0.0f;
    const float* r = F + (size_t)row * FD;
    #pragma unroll
    for (int c = tid; c < FD; c += 256) {
        float v = r[c];
        acc += v * v;
    }
    red[tid] = acc;
    __syncthreads();
    for (int s = 128; s > 0; s >>= 1) {
        if (tid < s) red[tid] += red[tid + s];
        __syncthreads();
    }
    if (tid == 0) sq[row] = red[0];
}

// ---------------------------------------------------------------------------
// Kernel 2: distance matrix via fp32 WMMA (V_WMMA_F32_16X16X4_F32).
// 8 waves per block, one 16x16 output tile per wave. 32x32 tiles total
// -> grid of 128 blocks. EXEC is all-ones everywhere (full blocks, no
// divergence around the WMMA).
//
// A fragment (16x4 f32): lane L holds row m=L&15, ks = kb + 2*(L>>4)+{0,1}.
// B fragment (4x16 f32) mirrors it with n=L&15 -> identical load pattern
// against the column tile's row of X (B = X^T slice).
// C/D: VGPR r holds M = 8*(L>>4)+r, N = L&15.
// ---------------------------------------------------------------------------
__global__ __launch_bounds__(256)
void gemm_dist_kernel(const float* __restrict__ F,
                      const float* __restrict__ sq,
                      float* __restrict__ dist) {
    const int wave = threadIdx.x >> 5;
    const int lane = threadIdx.x & 31;
    const int tile = blockIdx.x * 8 + wave;       // 0..1023
    const int tr   = tile >> 5;                   // tile row   0..31
    const int tc   = tile & 31;                   // tile col   0..31

    const int mrow = (tr << 4) + (lane & 15);     // A row of X
    const int ncol = (tc << 4) + (lane & 15);     // B row of X (column of C)
    const int koff = (lane >> 4) << 1;            // 0 or 2

    const float* arow = F + (size_t)mrow * FD + koff;
    const float* brow = F + (size_t)ncol * FD + koff;

    v8f acc = {0.f, 0.f, 0.f, 0.f, 0.f, 0.f, 0.f, 0.f};

    #pragma unroll 4
    for (int kb = 0; kb < FD; kb += 4) {
        v2f a = *(const v2f*)(arow + kb);
        v2f b = *(const v2f*)(brow + kb);
        acc = __builtin_amdgcn_wmma_f32_16x16x4_f32(
            /*neg_a=*/false, a, /*neg_b=*/false, b,
            /*c_mod=*/(short)0, acc, /*reuse_a=*/false, /*reuse_b=*/false);
    }

    const float sqn   = sq[ncol];
    const int   mbase = (tr << 4) + ((lane >> 4) << 3);
    #pragma unroll
    for (int r = 0; r < 8; ++r) {
        const int   mG = mbase + r;
        const float d2 = sq[mG] + sqn - 2.0f * acc[r];
        dist[(size_t)mG * NB + ncol] = sqrtf(fmaxf(d2, 1e-11f));
    }
}

// ---------------------------------------------------------------------------
// Kernel 3: zero the (sum,count) accumulators each launch (graph-replay safe).
// ---------------------------------------------------------------------------
__global__ void zero_acc_kernel(float* __restrict__ acc) {
    acc[0] = 0.0f;
    acc[1] = 0.0f;
}

// ---------------------------------------------------------------------------
// Kernel 4: semi-hard mining. One block (256 threads) per anchor i.
// dist row + labels staged in LDS; each thread owns 2 j's and scans k.
// ---------------------------------------------------------------------------
__global__ __launch_bounds__(256)
void mine_kernel(const float* __restrict__ dist,
                 const int* __restrict__ labels,
                 float* __restrict__ acc) {
    __shared__ float sdist[NB];
    __shared__ int   slab[NB];
    __shared__ float redS[256];
    __shared__ float redC[256];

    const int i   = blockIdx.x;
    const int tid = threadIdx.x;

    sdist[tid]       = dist[(size_t)i * NB + tid];
    sdist[tid + 256] = dist[(size_t)i * NB + tid + 256];
    slab[tid]        = labels[tid];
    slab[tid + 256]  = labels[tid + 256];
    __syncthreads();

    const int li = slab[i];

    const int   j0 = tid;
    const int   j1 = tid + 256;
    const float d0 = sdist[j0];
    const float d1 = sdist[j1];
    float m0 = BIGV, m1 = BIGV;

    for (int k = 0; k < NB; ++k) {
        const float dk  = sdist[k];          // LDS broadcast, conflict-free
        const bool  neg = (slab[k] != li);
        if (neg && dk > d0 && dk < d0 + MARGIN) m0 = fminf(m0, dk);
        if (neg && dk > d1 && dk < d1 + MARGIN) m1 = fminf(m1, dk);
    }

    float lsum = 0.0f, lcnt = 0.0f;
    const bool pos0 = (slab[j0] == li) && (j0 > i);
    const bool pos1 = (slab[j1] == li) && (j1 > i);
    if (pos0 && m0 < BIGV) { lsum += fmaxf(MARGIN - (d0 - m0), 0.0f); lcnt += 1.0f; }
    if (pos1 && m1 < BIGV) { lsum += fmaxf(MARGIN - (d1 - m1), 0.0f); lcnt += 1.0f; }

    redS[tid] = lsum;
    redC[tid] = lcnt;
    __syncthreads();
    for (int s = 128; s > 0; s >>= 1) {
        if (tid < s) { redS[tid] += redS[tid + s]; redC[tid] += redC[tid + s]; }
        __syncthreads();
    }
    if (tid == 0) {
        atomicAdd(&acc[0], redS[0]);   // global_atomic_add_f32
        atomicAdd(&acc[1], redC[0]);
    }
}

// ---------------------------------------------------------------------------
// Kernel 5: finalize  loss = sum / max(count, 1)
// ---------------------------------------------------------------------------
__global__ void finalize_kernel(const float* __restrict__ acc,
                                float* __restrict__ out) {
    out[0] = acc[0] / fmaxf(acc[1], 1.0f);
}

// ---------------------------------------------------------------------------
extern "C" void kernel_launch(void* const* d_in, const int* in_sizes, int n_in,
                              void* d_out, int out_size, void* d_ws, size_t ws_size,
                              hipStream_t stream) {
    const float* features = (const float*)d_in[0];   // 512 x 1024 fp32
    const int*   labels   = (const int*)d_in[1];     // 512 int
    // d_in[2] = epoch (unused: deterministic semi-hard branch)

    float* dist = (float*)d_ws;                      // 512*512 floats (1 MB)
    float* sq   = dist + (size_t)NB * NB;            // 512 floats
    float* acc  = sq + NB;                           // 2 floats (sum, count)
    float* out  = (float*)d_out;

    sqnorm_kernel   <<<NB, 256, 0, stream>>>(features, sq);
    gemm_dist_kernel<<<128, 256, 0, stream>>>(features, sq, dist);
    zero_acc_kernel <<<1, 1, 0, stream>>>(acc);
    mine_kernel     <<<NB, 256, 0, stream>>>(dist, labels, acc);
    finalize_kernel <<<1, 1, 0, stream>>>(acc, out);
    (void)in_sizes; (void)n_in; (void)out_size; (void)ws_size;
}